// CumulativeLinkLoss_30477087933027
// MI455X (gfx1250) — compile-verified
//
#include <hip/hip_runtime.h>
#include <hip/hip_bf16.h>

typedef __attribute__((ext_vector_type(2))) float v2f;
typedef __attribute__((ext_vector_type(8))) float v8f;

#define CLL_EPS 1e-15f

// Kernel 1: each wave processes 16-row tiles of y_pred (16x64 f32) using
// V_WMMA_F32_16X16X4_F32 as the masked row-sum engine. Per block, one f32
// partial of sum(-log(clip(masked_rowsum)) * w[idx]) is written to ws.
__global__ void __launch_bounds__(256)
cll_main_kernel(const float* __restrict__ y_pred,
                const int* __restrict__ y_true_lo,  // low dwords of int64 labels (stride 2)
                const float* __restrict__ cw,
                float* __restrict__ partial,
                int B) {
    const int lane         = threadIdx.x & 31;
    const int waveInBlock  = threadIdx.x >> 5;
    const int wavesPerBlk  = blockDim.x >> 5;
    const int waveId       = blockIdx.x * wavesPerBlk + waveInBlock;
    const int numWaves     = gridDim.x * wavesPerBlk;
    const int lo           = lane & 15;   // row within tile (A-matrix M)
    const int hi           = lane >> 4;   // selects K pair {2hi, 2hi+1}
    const int numTiles     = B >> 4;      // 16 rows per tile

    float acc = 0.0f;

    for (int tile = waveId; tile < numTiles; tile += numWaves) {
        const int row = (tile << 4) + lo;
        // labels are in [0,64): only the low dword of the int64 matters
        const int idx = y_true_lo[row << 1];
        const float wt = cw[idx];
        const float* rowp = y_pred + ((long long)row << 6);

        // speculative prefetch of the next grid-stride tile (OOB is dropped)
        __builtin_prefetch(y_pred + ((((long long)(tile + numWaves) << 4) + lo) << 6), 0, 0);

        // Single unsigned range test replaces (idx==0 ? col==0 : col>=idx):
        // include col iff (unsigned)(col - idx) <= range
        const unsigned range = (idx == 0) ? 0u : (unsigned)(63 - idx);
        const int base = (hi << 1) - idx;   // (col0 - idx) minus the 4*kb term

        v8f c = {};                    // 16x16 f32 accumulator (rowsum replicated over N)
        v2f b; b.x = 1.0f; b.y = 1.0f; // B = 4x16 all-ones

        #pragma unroll
        for (int kb = 0; kb < 16; ++kb) {
            const int col0 = (kb << 2) + (hi << 1);
            const float x0 = rowp[col0];
            const float x1 = rowp[col0 + 1];
            const int d0 = base + (kb << 2);
            const bool t0 = (unsigned)d0       <= range;
            const bool t1 = (unsigned)(d0 + 1) <= range;
            v2f a;
            a.x = t0 ? x0 : 0.0f;
            a.y = t1 ? x1 : 0.0f;
            // D = A(16x4, f32) * ones(4x16) + C  -> per-row partial sums, f32 RNE
            c = __builtin_amdgcn_wmma_f32_16x16x4_f32(
                    false, a, false, b, (short)0, c, false, false);
        }

        // Owner lanes (16 unique rows): lanes 0-7 -> rows 0-7 (c[lo]),
        // lanes 24-31 -> rows 8-15 (c[lo-8]). Each owner lane loaded its own
        // row's idx/weight above (row == tileBase + lo).
        const bool owner = (hi == 0) ? (lo < 8) : (lo >= 8);
        const float s   = c[lo & 7];
        const float lik = fminf(fmaxf(s, CLL_EPS), 1.0f - CLL_EPS);
        const float nll = -logf(lik) * wt;
        acc += owner ? nll : 0.0f;
    }

    // wave32 reduction
    #pragma unroll
    for (int off = 16; off > 0; off >>= 1)
        acc += __shfl_xor(acc, off, 32);

    __shared__ float sdata[8];
    if (lane == 0) sdata[waveInBlock] = acc;
    __syncthreads();
    if (threadIdx.x == 0) {
        float t = 0.0f;
        for (int i = 0; i < wavesPerBlk; ++i) t += sdata[i];
        partial[blockIdx.x] = t;
    }
}

// Kernel 2: deterministic single-block reduction of per-block partials.
__global__ void __launch_bounds__(256)
cll_reduce_kernel(const float* __restrict__ partial, int n,
                  float* __restrict__ out, float invB) {
    float acc = 0.0f;
    for (int i = threadIdx.x; i < n; i += blockDim.x)
        acc += partial[i];
    #pragma unroll
    for (int off = 16; off > 0; off >>= 1)
        acc += __shfl_xor(acc, off, 32);

    __shared__ float sdata[8];
    const int lane = threadIdx.x & 31;
    const int wv   = threadIdx.x >> 5;
    if (lane == 0) sdata[wv] = acc;
    __syncthreads();
    if (threadIdx.x == 0) {
        float t = 0.0f;
        const int nw = blockDim.x >> 5;
        for (int i = 0; i < nw; ++i) t += sdata[i];
        out[0] = t * invB;
    }
}

extern "C" void kernel_launch(void* const* d_in, const int* in_sizes, int n_in,
                              void* d_out, int out_size, void* d_ws, size_t ws_size,
                              hipStream_t stream) {
    const float* y_pred    = (const float*)d_in[0];
    const int*   y_true_lo = (const int*)d_in[1];   // int64 labels, low dword at stride 2
    const float* cw        = (const float*)d_in[2];
    float* out = (float*)d_out;
    float* ws  = (float*)d_ws;

    const int B = in_sizes[0] / 64;   // rows; C == 64 per reference

    const int blocks  = 2048;         // 8 KB of partials in d_ws
    const int threads = 256;          // 8 waves/block

    cll_main_kernel<<<blocks, threads, 0, stream>>>(y_pred, y_true_lo, cw, ws, B);

    const float invB = 1.0f / (float)B;
    cll_reduce_kernel<<<1, threads, 0, stream>>>(ws, blocks, out, invB);
}